// GCN_70789650972705
// MI455X (gfx1250) — compile-verified
//
#include <hip/hip_runtime.h>
#include <hip/hip_bf16.h>
#include <math.h>

// ---------------------------------------------------------------------------
// GCN inference for MI455X (gfx1250, wave32).
//   layer1: h1 = x @ W1 (WMMA f32 16x16x4), aggregate via L2 f32 atomics,
//           fused selfloop+bias+relu
//   layer2: h2 = h @ W2 (WMMA f32, N padded 48), aggregate, fused
//           selfloop+bias+log_softmax
// ---------------------------------------------------------------------------

typedef float v2f __attribute__((ext_vector_type(2)));
typedef float v8f __attribute__((ext_vector_type(8)));

#define N_FEAT  512
#define N_HID   16
#define N_CLASS 40

// ---------------- degree / norm ----------------

__global__ void gcn_init_deg(float* deg, int n) {
    int i = blockIdx.x * blockDim.x + threadIdx.x;
    if (i < n) deg[i] = 1.0f;               // self-loop contributes 1 to every node
}

__global__ void gcn_degree(const int* __restrict__ dst, float* __restrict__ deg, int e) {
    int i = blockIdx.x * blockDim.x + threadIdx.x;
    if (i < e) atomicAdd(&deg[dst[i]], 1.0f);
}

__global__ void gcn_dinv(float* deg, int n) {   // in place: deg -> 1/sqrt(deg)
    int i = blockIdx.x * blockDim.x + threadIdx.x;
    if (i < n) deg[i] = rsqrtf(deg[i]);     // deg >= 1 always (self loop)
}

// ---------------- GEMM1: h1 = x @ W1  (M=100000, K=512, N=16) ----------------
// One wave per 16-row tile. f32 WMMA 16x16x4, 128 k-steps.

__global__ void gcn_gemm1(const float* __restrict__ x, const float* __restrict__ W,
                          float* __restrict__ h1, int mtiles) {
    int wid  = (blockIdx.x * blockDim.x + threadIdx.x) >> 5;   // global wave id
    int lane = threadIdx.x & 31;
    if (wid >= mtiles) return;                                 // wave-uniform: EXEC stays full

    const int m0   = wid * 16;
    const int half = lane >> 4;          // 0: lanes 0-15, 1: lanes 16-31
    const int l15  = lane & 15;
    const int koff = half * 2;           // A frag: K = koff, koff+1

    const float* xr = x + (size_t)(m0 + l15) * N_FEAT + koff;  // A row pointer
    const float* wc = W + (size_t)koff * N_HID + l15;          // B col pointer

    v8f c = {};
#pragma unroll 4
    for (int k = 0; k < N_FEAT; k += 4) {
        v2f a = *(const v2f*)(xr + k);           // 8B-aligned (k, koff even)
        v2f b;
        b.x = wc[k * N_HID];                     // W[(k+koff)  ][n]
        b.y = wc[k * N_HID + N_HID];             // W[(k+koff+1)][n]
        c = __builtin_amdgcn_wmma_f32_16x16x4_f32(false, a, false, b,
                                                  (short)0, c, false, false);
    }
    // C/D layout: VGPR r -> row m0 + r + 8*half, col = l15
#pragma unroll
    for (int r = 0; r < 8; ++r) {
        int m = m0 + r + (half << 3);
        h1[(size_t)m * N_HID + l15] = c[r];
    }
}

// ---------------- edge aggregation, layer 1 (F = 16) ----------------

__global__ void gcn_agg16(const int* __restrict__ src, const int* __restrict__ dst,
                          const float* __restrict__ dinv, const float* __restrict__ h,
                          float* __restrict__ agg, unsigned total) {
    unsigned idx = blockIdx.x * blockDim.x + threadIdx.x;
    if (idx >= total) return;
    unsigned e = idx >> 4;
    unsigned f = idx & 15u;
    int s = src[e], d = dst[e];
    float v = dinv[s] * dinv[d] * h[(size_t)s * N_HID + f];
    atomicAdd(&agg[(size_t)d * N_HID + f], v);   // global_atomic_add_f32, L2-resident
}

// ---------------- fused selfloop + bias + relu ----------------

__global__ void gcn_relu(const float* __restrict__ agg, const float* __restrict__ h1,
                         const float* __restrict__ dinv, const float* __restrict__ b1,
                         float* __restrict__ hout, unsigned total) {
    unsigned idx = blockIdx.x * blockDim.x + threadIdx.x;
    if (idx >= total) return;
    unsigned i = idx >> 4;
    unsigned f = idx & 15u;
    float di = dinv[i];
    float v  = agg[idx] + di * di * h1[idx] + b1[f];
    hout[idx] = fmaxf(v, 0.0f);
}

// ---------------- GEMM2: h2 = h @ W2 (M=100000, K=16, N=40 padded to 48) ------

__global__ void gcn_gemm2(const float* __restrict__ h, const float* __restrict__ W2,
                          float* __restrict__ h2, int mtiles) {
    int wid  = (blockIdx.x * blockDim.x + threadIdx.x) >> 5;
    int lane = threadIdx.x & 31;
    if (wid >= mtiles * 3) return;                 // wave-uniform exit

    int mtile = wid / 3;
    int ntile = wid - mtile * 3;

    const int m0   = mtile * 16;
    const int half = lane >> 4;
    const int l15  = lane & 15;
    const int koff = half * 2;
    const int col  = ntile * 16 + l15;             // 0..47, valid < 40

    const float* hr = h + (size_t)(m0 + l15) * N_HID + koff;

    v8f c = {};
#pragma unroll
    for (int k = 0; k < N_HID; k += 4) {
        v2f a = *(const v2f*)(hr + k);
        int kk = k + koff;
        v2f b;
        b.x = (col < N_CLASS) ? W2[kk * N_CLASS + col]       : 0.0f;
        b.y = (col < N_CLASS) ? W2[(kk + 1) * N_CLASS + col] : 0.0f;
        c = __builtin_amdgcn_wmma_f32_16x16x4_f32(false, a, false, b,
                                                  (short)0, c, false, false);
    }
    if (col < N_CLASS) {
#pragma unroll
        for (int r = 0; r < 8; ++r) {
            int m = m0 + r + (half << 3);
            h2[(size_t)m * N_CLASS + col] = c[r];
        }
    }
}

// ---------------- edge aggregation, layer 2 (F = 40) ----------------

__global__ void gcn_agg40(const int* __restrict__ src, const int* __restrict__ dst,
                          const float* __restrict__ dinv, const float* __restrict__ h,
                          float* __restrict__ agg, unsigned total) {
    unsigned idx = blockIdx.x * blockDim.x + threadIdx.x;
    if (idx >= total) return;
    unsigned e = idx / 40u;                         // mul-hi, cheap
    unsigned f = idx - e * 40u;
    int s = src[e], d = dst[e];
    float v = dinv[s] * dinv[d] * h[(size_t)s * N_CLASS + f];
    atomicAdd(&agg[(size_t)d * N_CLASS + f], v);
}

// ---------------- fused selfloop + bias + log_softmax ----------------

__global__ void gcn_logsoftmax(const float* __restrict__ agg, const float* __restrict__ h2,
                               const float* __restrict__ dinv, const float* __restrict__ b2,
                               float* __restrict__ out, int n) {
    int i = blockIdx.x * blockDim.x + threadIdx.x;
    if (i >= n) return;
    float di = dinv[i];
    float d2 = di * di;
    float v[N_CLASS];
    float mx = -INFINITY;
#pragma unroll
    for (int c = 0; c < N_CLASS; ++c) {
        float t = agg[(size_t)i * N_CLASS + c] + d2 * h2[(size_t)i * N_CLASS + c] + b2[c];
        v[c] = t;
        mx = fmaxf(mx, t);
    }
    float s = 0.0f;
#pragma unroll
    for (int c = 0; c < N_CLASS; ++c) s += expf(v[c] - mx);
    float lse = mx + logf(s);
#pragma unroll
    for (int c = 0; c < N_CLASS; ++c) out[(size_t)i * N_CLASS + c] = v[c] - lse;
}

// ---------------------------------------------------------------------------

extern "C" void kernel_launch(void* const* d_in, const int* in_sizes, int n_in,
                              void* d_out, int out_size, void* d_ws, size_t ws_size,
                              hipStream_t stream) {
    const float* x  = (const float*)d_in[0];
    const int*   ei = (const int*)  d_in[1];
    const float* W1 = (const float*)d_in[2];
    const float* b1 = (const float*)d_in[3];
    const float* W2 = (const float*)d_in[4];
    const float* b2 = (const float*)d_in[5];

    const int n = in_sizes[0] / N_FEAT;    // 100000 (divisible by 16)
    const int e = in_sizes[1] / 2;         // 3200000
    const int* src = ei;                   // edge_index[0]
    const int* dst = ei + e;               // edge_index[1]

    // workspace layout (floats)
    float* ws    = (float*)d_ws;
    float* dinv  = ws;                              // n
    float* h1    = dinv + n;                        // n*16
    float* agg1  = h1   + (size_t)n * N_HID;        // n*16
    float* hrelu = agg1 + (size_t)n * N_HID;        // n*16
    float* h2    = hrelu + (size_t)n * N_HID;       // n*40
    float* agg2  = h2   + (size_t)n * N_CLASS;      // n*40

    hipMemsetAsync(agg1, 0, (size_t)n * N_HID   * sizeof(float), stream);
    hipMemsetAsync(agg2, 0, (size_t)n * N_CLASS * sizeof(float), stream);

    const int B = 256;

    // normalization: deg (with self loop) -> dinv
    gcn_init_deg<<<(n + B - 1) / B, B, 0, stream>>>(dinv, n);
    gcn_degree  <<<(e + B - 1) / B, B, 0, stream>>>(dst, dinv, e);
    gcn_dinv    <<<(n + B - 1) / B, B, 0, stream>>>(dinv, n);

    // layer 1
    const int mtiles = n / 16;                                  // 6250
    gcn_gemm1<<<(mtiles * 32 + B - 1) / B, B, 0, stream>>>(x, W1, h1, mtiles);

    unsigned tot1 = (unsigned)e * N_HID;                        // 51.2M
    gcn_agg16<<<(tot1 + B - 1) / B, B, 0, stream>>>(src, dst, dinv, h1, agg1, tot1);

    unsigned totn1 = (unsigned)n * N_HID;
    gcn_relu<<<(totn1 + B - 1) / B, B, 0, stream>>>(agg1, h1, dinv, b1, hrelu, totn1);

    // layer 2
    gcn_gemm2<<<(mtiles * 3 * 32 + B - 1) / B, B, 0, stream>>>(hrelu, W2, h2, mtiles);

    unsigned tot2 = (unsigned)e * N_CLASS;                      // 128M
    gcn_agg40<<<(tot2 + B - 1) / B, B, 0, stream>>>(src, dst, dinv, h2, agg2, tot2);

    gcn_logsoftmax<<<(n + B - 1) / B, B, 0, stream>>>(agg2, h2, dinv, b2,
                                                      (float*)d_out, n);
}